// NonLinearLayer_64304250356459
// MI455X (gfx1250) — compile-verified
//
#include <hip/hip_runtime.h>
#include <hip/hip_bf16.h>
#include <math.h>

#define NUM_ELMT 64
#define IN_DIM   16
#define BOUND    50.0f

// mesh constants: r=1.2, one_step = 10/(1.2^32-1)
// COEF = (r-1)/one_step = 0.02*(1.2^32-1);  INV_LOG2R = 1/log2(1.2)
#define MESH_COEF   6.81643783743337f
#define INV_LOG2R   3.8017840169239308f

// ---------------------------------------------------------------------------
// gfx1250 async global->LDS staging (compile-guarded; falls back to plain loads)
// ---------------------------------------------------------------------------
#if defined(__gfx1250__) && __has_builtin(__builtin_amdgcn_global_load_async_to_lds_b32)
#define USE_ASYNC_LDS 1
#else
#define USE_ASYNC_LDS 0
#endif

#if USE_ASYNC_LDS
typedef int v4i __attribute__((ext_vector_type(4)));
typedef __attribute__((address_space(1))) int as1_int;
typedef __attribute__((address_space(3))) int as3_int;
typedef __attribute__((address_space(1))) v4i as1_v4i;
typedef __attribute__((address_space(3))) v4i as3_v4i;

__device__ __forceinline__ void async_cp_b32(const float* g, float* l) {
  __builtin_amdgcn_global_load_async_to_lds_b32((as1_int*)(g), (as3_int*)(l), 0, 0);
}
#if __has_builtin(__builtin_amdgcn_global_load_async_to_lds_b128)
__device__ __forceinline__ void async_cp_b128(const float4* g, float4* l) {
  __builtin_amdgcn_global_load_async_to_lds_b128((as1_v4i*)(g), (as3_v4i*)(l), 0, 0);
}
#else
__device__ __forceinline__ void async_cp_b128(const float4* g, float4* l) {
  const float* gf = (const float*)g;
  float* lf = (float*)l;
  async_cp_b32(gf + 0, lf + 0); async_cp_b32(gf + 1, lf + 1);
  async_cp_b32(gf + 2, lf + 2); async_cp_b32(gf + 3, lf + 3);
}
#endif
__device__ __forceinline__ void async_wait0() {
#if __has_builtin(__builtin_amdgcn_s_wait_asynccnt)
  __builtin_amdgcn_s_wait_asynccnt(0);
#else
  asm volatile("s_wait_asynccnt 0" ::: "memory");
#endif
}
#endif // USE_ASYNC_LDS

// ---------------------------------------------------------------------------
// Kernel 1: build packed per-(dim,cell) table {v1, slope=(v2-v1)/h, F_pre, 0}
// into ws as float4[d*64 + cell].  16 threads (one per dim); recomputed each call.
// ---------------------------------------------------------------------------
__global__ void nl_build_tables(const float* __restrict__ log_p,
                                const float* __restrict__ elmt_size,
                                float4* __restrict__ pk) {   // 16*64 float4
  int d = threadIdx.x;
  if (d >= IN_DIM) return;

  float pdfc[NUM_ELMT + 1];

  // normalization: sum over exp(log_p)*(es[i]+es[i+1])/2
  float sum = 0.0f;
  for (int i = 0; i < NUM_ELMT - 1; ++i) {
    float e = expf(log_p[i * IN_DIM + d]);
    pdfc[i + 1] = e;                       // stash exp temporarily
    sum += e * (elmt_size[i] + elmt_size[i + 1]) * 0.5f;
  }
  float scale = (1.0f - elmt_size[0]) / sum;

  pdfc[0] = 1.0f;
  for (int i = 1; i < NUM_ELMT; ++i) pdfc[i] *= scale;
  pdfc[NUM_ELMT] = 1.0f;

  // F_ref[i] = sum_{j<i} (pdf[j]+pdf[j+1])/2 * es[j]; pack {v1, slope, F}
  float acc = 0.0f;
  for (int i = 0; i < NUM_ELMT; ++i) {
    float v1 = pdfc[i];
    float v2 = pdfc[i + 1];
    float h  = elmt_size[i];
    float4 t;
    t.x = v1;
    t.y = (v2 - v1) / h;   // slope: kills the per-element fp32 divide
    t.z = acc;             // F_pre
    t.w = 0.0f;
    pk[d * NUM_ELMT + i] = t;
    acc += (v1 + v2) * 0.5f * h;
  }
}

// ---------------------------------------------------------------------------
// Kernel 2: main elementwise transform. One thread per point (16 dims).
// Analytic searchsorted seed (one v_log_f32) + exact +-1 LDS fixup.
// ---------------------------------------------------------------------------
__global__ void __launch_bounds__(256)
nl_transform(const float* __restrict__ x,
             const float* __restrict__ pdj,
             const float* __restrict__ sldj,
             const float* __restrict__ mesh_norm,   // 65
             const float4* __restrict__ pk,         // 16*64 (ws)
             float* __restrict__ x_out,
             float* __restrict__ pdj_out,
             float* __restrict__ sldj_out,
             int n) {
  __shared__ float  s_mesh[NUM_ELMT + 1];            // 65 floats
  __shared__ float4 s_pk[IN_DIM * NUM_ELMT];         // 1024 float4 = 16 KB

  const int tid  = threadIdx.x;
  const int nthr = blockDim.x;

#if USE_ASYNC_LDS
  for (int i = tid; i < IN_DIM * NUM_ELMT; i += nthr) async_cp_b128(&pk[i], &s_pk[i]);
  for (int i = tid; i < NUM_ELMT + 1;      i += nthr) async_cp_b32(&mesh_norm[i], &s_mesh[i]);
  async_wait0();
#else
  for (int i = tid; i < IN_DIM * NUM_ELMT; i += nthr) s_pk[i] = pk[i];
  for (int i = tid; i < NUM_ELMT + 1;      i += nthr) s_mesh[i] = mesh_norm[i];
#endif
  __syncthreads();

  const int p = blockIdx.x * nthr + tid;
  if (p >= n) return;

  // 4x global_load_b128, coalesced per wave (512 B/wave/load)
  float xr[IN_DIM];
  const float4* xp = reinterpret_cast<const float4*>(x + (size_t)p * IN_DIM);
#pragma unroll
  for (int q = 0; q < 4; ++q) {
    float4 v = xp[q];
    xr[q * 4 + 0] = v.x; xr[q * 4 + 1] = v.y;
    xr[q * 4 + 2] = v.z; xr[q * 4 + 3] = v.w;
  }

  float prod = 1.0f;
  float slog = 0.0f;

#pragma unroll
  for (int d = 0; d < IN_DIM; ++d) {
    float xv = xr[d];
    float xn = (xv + BOUND) * (1.0f / (2.0f * BOUND));

    // ---- analytic seed for searchsorted(mesh_norm, xn, 'left') ----
    // mesh is geometric: |mesh_orig(j)| = (r^|j|-1)/(r-1)*one_step, j = k-32
    // and |xn - 0.5| * 2*BOUND == |x| exactly, so seed from |x| directly.
    float jf = __log2f(fmaf(fabsf(xv), MESH_COEF, 1.0f)) * INV_LOG2R;
    int  ji  = (int)jf;                       // floor (jf >= 0)
    int  k   = (xv >= 0.0f) ? (33 + ji) : (32 - ji);
    k = k < 0 ? 0 : (k > NUM_ELMT + 1 ? NUM_ELMT + 1 : k);

    // ---- exact branch-free fixup (seed is within +-1 cell; inc then dec) ----
    {
      float mk = s_mesh[k > NUM_ELMT ? NUM_ELMT : k];
      k += ((k < NUM_ELMT + 1) && (mk < xn)) ? 1 : 0;
      mk = s_mesh[k > 0 ? k - 1 : 0];
      k -= ((k > 0) && (mk >= xn)) ? 1 : 0;
    }

    bool cover = (k != 0) && (k != NUM_ELMT + 1);
    int kc  = k < 1 ? 1 : (k > NUM_ELMT ? NUM_ELMT : k);
    int km1 = kc - 1;

    // one ds_load_b128 gather; dim-major layout spreads banks by 4*km1
    float4 t4 = s_pk[(d << 6) + km1];         // {v1, slope, F_pre, 0}
    float xm  = xn - s_mesh[km1];

    float y  = cover ? (t4.z + xm * (xm * (0.5f * t4.y) + t4.x)) : xn;
    float ad = cover ? (xm * t4.y + t4.x) : 1.0f;

    xr[d] = fmaf(y, 2.0f * BOUND, -BOUND);
    prod *= ad;
    slog += __logf(ad);
  }

  float4* op = reinterpret_cast<float4*>(x_out + (size_t)p * IN_DIM);
#pragma unroll
  for (int q = 0; q < 4; ++q) {
    float4 v;
    v.x = xr[q * 4 + 0]; v.y = xr[q * 4 + 1];
    v.z = xr[q * 4 + 2]; v.w = xr[q * 4 + 3];
    op[q] = v;
  }
  pdj_out[p]  = pdj[p] * prod;
  sldj_out[p] = sldj[p] + slog;
}

// ---------------------------------------------------------------------------
// Launch
// ---------------------------------------------------------------------------
extern "C" void kernel_launch(void* const* d_in, const int* in_sizes, int n_in,
                              void* d_out, int out_size, void* d_ws, size_t ws_size,
                              hipStream_t stream) {
  const float* x         = (const float*)d_in[0];
  const float* pdj       = (const float*)d_in[1];
  const float* sldj      = (const float*)d_in[2];
  const float* log_p     = (const float*)d_in[3];
  const float* mesh_norm = (const float*)d_in[4];
  const float* elmt_size = (const float*)d_in[5];

  const int n = in_sizes[0] / IN_DIM;

  float4* ws_pk = (float4*)d_ws;               // 16*64 float4 = 16 KB

  float* x_out    = (float*)d_out;
  float* pdj_out  = x_out + (size_t)n * IN_DIM;
  float* sldj_out = pdj_out + n;

  nl_build_tables<<<1, IN_DIM, 0, stream>>>(log_p, elmt_size, ws_pk);

  const int block = 256;
  const int grid  = (n + block - 1) / block;
  nl_transform<<<grid, block, 0, stream>>>(x, pdj, sldj, mesh_norm, ws_pk,
                                           x_out, pdj_out, sldj_out, n);
}